// AugmentedConv3D_59571196395716
// MI455X (gfx1250) — compile-verified
//
#include <hip/hip_runtime.h>

typedef __attribute__((ext_vector_type(2))) float v2f;
typedef __attribute__((ext_vector_type(8))) float v8f;

#define PP   4096      // 16*16*16 spatial points
#define CIN  32
#define KTOT 864       // 32 * 27
#define KC   16        // K chunk staged in LDS per iteration
#define WSTRIDE 20     // padded row stride (floats): 80B rows -> 16B aligned, bank-spread

// ---------------------------------------------------------------------------
// Kernel 1: fused conv3d for Wc (48 ch) and Wqkv (80 ch) as implicit GEMM.
// One wave per (b, z, y) x-row tile of 16 points. M = 128 in 8 WMMA tiles.
// Weights staged per-block into LDS via gfx1250 async-to-LDS copies; A
// fragments then come from conflict-free ds_load_b64.
// ---------------------------------------------------------------------------
__global__ __launch_bounds__(256) void conv_qkv_wmma(
    const float* __restrict__ x,
    const float* __restrict__ Wc,   const float* __restrict__ bc,
    const float* __restrict__ Wqkv, const float* __restrict__ bqkv,
    float* __restrict__ out, float* __restrict__ qkvws)
{
    __shared__ float wlds[128 * WSTRIDE];   // 10 KB weight panel

    const int wid  = (blockIdx.x * blockDim.x + threadIdx.x) >> 5;  // 0..511
    const int lane = threadIdx.x & 31;
    const int col  = lane & 15;     // N column within tile (x position)
    const int hi   = lane >> 4;     // lane half selects K pair

    const int b  = wid >> 8;
    const int zy = wid & 255;
    const int z  = zy >> 4;
    const int y  = zy & 15;

    // staging role: each thread owns (row, 8-float half-chunk)
    const int srow  = threadIdx.x >> 1;          // 0..127
    const int skoff = (threadIdx.x & 1) * 8;     // 0 or 8
    const float* wsrc = ((srow < 48) ? (Wc + (size_t)srow * KTOT)
                                     : (Wqkv + (size_t)(srow - 48) * KTOT)) + skoff;
    // flat LDS address: low 32 bits of the flat pointer are the LDS byte offset
    const unsigned slds = (unsigned)(uintptr_t)(void*)&wlds[srow * WSTRIDE + skoff];

    // C/D layout: VGPR r <-> M = r (lanes 0-15) or M = 8+r (lanes 16-31)
    v8f acc[8];
#pragma unroll
    for (int mt = 0; mt < 8; ++mt)
#pragma unroll
        for (int r = 0; r < 8; ++r) {
            const int m = mt * 16 + hi * 8 + r;
            acc[mt][r] = (m < 48) ? bc[m] : bqkv[m - 48];
        }

    const float* xb = x + (size_t)b * CIN * PP;

    for (int kc = 0; kc < KTOT; kc += KC) {
        __syncthreads();   // previous chunk's LDS readers are done
        {
            const unsigned long long ga =
                (unsigned long long)(uintptr_t)(const void*)(wsrc + kc);
            // async copy 2 x 16B per thread: 128 rows x 16 K-values
            asm volatile(
                "global_load_async_to_lds_b128 %0, %1, off\n\t"
                "global_load_async_to_lds_b128 %0, %1, off offset:16"
                :: "v"(slds), "v"(ga) : "memory");
        }
        // prefetch next chunk's weights into L2 while this one is consumed
        if (kc + KC < KTOT) __builtin_prefetch(wsrc + kc + KC, 0, 0);
        asm volatile("s_wait_asynccnt 0x0" ::: "memory");
        __syncthreads();   // panel visible to all waves

#pragma unroll 4
        for (int k0c = 0; k0c < KC; k0c += 4) {
            // B tile (im2col patch): lane = N (position), VGPR r + half -> K
            float bv[2];
#pragma unroll
            for (int r = 0; r < 2; ++r) {
                const int kk = kc + k0c + hi * 2 + r;
                const int ci = kk / 27;
                int rem = kk - ci * 27;
                const int dz = rem / 9;  rem -= dz * 9;
                const int dy = rem / 3;
                const int dx = rem - dy * 3;
                const int zz = z + dz - 1, yy = y + dy - 1, xx = col + dx - 1;
                const bool ok = ((unsigned)zz < 16u) & ((unsigned)yy < 16u) &
                                ((unsigned)xx < 16u);
                bv[r] = ok ? xb[ci * PP + zz * 256 + yy * 16 + xx] : 0.0f;
            }
            const v2f B = { bv[0], bv[1] };

#pragma unroll
            for (int mt = 0; mt < 8; ++mt) {
                // A fragment from LDS: lane = M row, VGPR r + half -> K pair
                const v2f A = *(const v2f*)&wlds[(mt * 16 + col) * WSTRIDE +
                                                 k0c + hi * 2];
                acc[mt] = __builtin_amdgcn_wmma_f32_16x16x4_f32(
                    false, A, false, B, (short)0, acc[mt], false, false);
            }
        }
    }

    const int p = z * 256 + y * 16 + col;
#pragma unroll
    for (int mt = 0; mt < 8; ++mt)
#pragma unroll
        for (int r = 0; r < 8; ++r) {
            const int m = mt * 16 + hi * 8 + r;
            const float v = acc[mt][r];
            if (m < 48) out[((size_t)b * 64 + m) * PP + p] = v;
            else        qkvws[((size_t)b * 80 + (m - 48)) * PP + p] = v;
        }
}

// ---------------------------------------------------------------------------
// Kernel 2: flash attention. One wave per (b, head, 16-query tile).
// S^T tile (rows = key m, cols = query n) = K_chunk x Q_chunk via 2 fp32
// WMMAs (dkh = 8 = 2 x K4). Online softmax over m: in-lane reduce over the
// 8 accumulator VGPRs + shfl_xor(16) across lane halves (wave32).
// ---------------------------------------------------------------------------
__global__ __launch_bounds__(256) void attn_wmma(
    const float* __restrict__ qkv, float* __restrict__ attnws)
{
    const int wid  = (blockIdx.x * blockDim.x + threadIdx.x) >> 5;
    const int lane = threadIdx.x & 31;
    const int col  = lane & 15;
    const int hi   = lane >> 4;
    if (wid >= 2048) return;

    const int nt = wid & 255;
    const int h  = (wid >> 8) & 3;
    const int b  = wid >> 10;
    const int n0 = nt * 16;

    const float scale = 0.35355339059327373f;   // dkh^-0.5 = 8^-0.5
    const float* qb = qkv + ((size_t)b * 80 +      h * 8) * PP;
    const float* kb = qkv + ((size_t)b * 80 + 32 + h * 8) * PP;
    const float* vb = qkv + ((size_t)b * 80 + 64 + h * 4) * PP;

    // Q as B-matrix tiles: lane = n column, VGPR r + half -> channel
    const v2f q0 = { qb[(hi * 2 + 0) * PP + n0 + col] * scale,
                     qb[(hi * 2 + 1) * PP + n0 + col] * scale };
    const v2f q1 = { qb[(4 + hi * 2 + 0) * PP + n0 + col] * scale,
                     qb[(4 + hi * 2 + 1) * PP + n0 + col] * scale };

    float M = -1.0e30f, l = 0.0f;
    float o0 = 0.f, o1 = 0.f, o2 = 0.f, o3 = 0.f;

    for (int mb = 0; mb < 256; ++mb) {
        const int m0 = mb * 16;
        // K as A-matrix tiles: lane = m row, VGPR r + half -> channel
        const v2f a0 = { kb[(hi * 2 + 0) * PP + m0 + col],
                         kb[(hi * 2 + 1) * PP + m0 + col] };
        const v2f a1 = { kb[(4 + hi * 2 + 0) * PP + m0 + col],
                         kb[(4 + hi * 2 + 1) * PP + m0 + col] };

        v8f s = { 0.f, 0.f, 0.f, 0.f, 0.f, 0.f, 0.f, 0.f };
        s = __builtin_amdgcn_wmma_f32_16x16x4_f32(
                false, a0, false, q0, (short)0, s, false, false);
        s = __builtin_amdgcn_wmma_f32_16x16x4_f32(
                false, a1, false, q1, (short)0, s, false, false);

        // ---- online softmax over m (rows of S^T) for column n = n0+col ----
        float tm = s[0];
#pragma unroll
        for (int r = 1; r < 8; ++r) tm = fmaxf(tm, s[r]);
        tm = fmaxf(tm, __shfl_xor(tm, 16, 32));
        const float Mn = fmaxf(M, tm);
        const float rs = __expf(M - Mn);

        float pr[8], ps = 0.f;
#pragma unroll
        for (int r = 0; r < 8; ++r) { pr[r] = __expf(s[r] - Mn); ps += pr[r]; }
        ps += __shfl_xor(ps, 16, 32);
        l = l * rs + ps;
        M = Mn;

        o0 *= rs; o1 *= rs; o2 *= rs; o3 *= rs;
#pragma unroll
        for (int r = 0; r < 8; ++r) {
            const int m = m0 + hi * 8 + r;
            const float p = pr[r];
            o0 += p * vb[0 * PP + m];
            o1 += p * vb[1 * PP + m];
            o2 += p * vb[2 * PP + m];
            o3 += p * vb[3 * PP + m];
        }
    }

    // combine the two lane halves (each covered half of the m rows)
    o0 += __shfl_xor(o0, 16, 32);
    o1 += __shfl_xor(o1, 16, 32);
    o2 += __shfl_xor(o2, 16, 32);
    o3 += __shfl_xor(o3, 16, 32);

    if (hi == 0) {
        const float inv = 1.0f / l;
        const int n = n0 + col;
        float oc[4] = { o0 * inv, o1 * inv, o2 * inv, o3 * inv };
#pragma unroll
        for (int c = 0; c < 4; ++c) {
            // reference reshapes (P, dvh) -> (dvh, D, H, W) without transpose
            const int L  = n * 4 + c;
            const int c2 = L >> 12;
            const int p2 = L & 4095;
            attnws[((size_t)b * 16 + h * 4 + c2) * PP + p2] = oc[c];
        }
    }
}

// ---------------------------------------------------------------------------
// Kernel 3: 1x1x1 output projection Wo (16x16) + bias -> out channels 48..63
// ---------------------------------------------------------------------------
__global__ __launch_bounds__(256) void proj_wo(
    const float* __restrict__ attnws, const float* __restrict__ Wo,
    const float* __restrict__ bo, float* __restrict__ out)
{
    const int idx = blockIdx.x * blockDim.x + threadIdx.x;
    if (idx >= 2 * PP) return;
    const int b = idx >> 12;
    const int p = idx & 4095;

    float a[16];
#pragma unroll
    for (int c = 0; c < 16; ++c) a[c] = attnws[((size_t)b * 16 + c) * PP + p];

#pragma unroll
    for (int o = 0; o < 16; ++o) {
        float s = bo[o];
#pragma unroll
        for (int c = 0; c < 16; ++c) s += Wo[o * 16 + c] * a[c];
        out[((size_t)b * 64 + 48 + o) * PP + p] = s;
    }
}

// ---------------------------------------------------------------------------
extern "C" void kernel_launch(void* const* d_in, const int* in_sizes, int n_in,
                              void* d_out, int out_size, void* d_ws, size_t ws_size,
                              hipStream_t stream)
{
    const float* x    = (const float*)d_in[0];
    const float* Wc   = (const float*)d_in[1];
    const float* bc   = (const float*)d_in[2];
    const float* Wqkv = (const float*)d_in[3];
    const float* bqkv = (const float*)d_in[4];
    const float* Wo   = (const float*)d_in[5];
    const float* bo   = (const float*)d_in[6];
    float* out = (float*)d_out;

    float* qkvws  = (float*)d_ws;                 // 2*80*4096 floats
    float* attnws = qkvws + (size_t)2 * 80 * PP;  // 2*16*4096 floats

    // 512 waves (one per (b,z,y) row tile), 8 waves/block -> exactly 64 blocks
    conv_qkv_wmma<<<64, 256, 0, stream>>>(x, Wc, bc, Wqkv, bqkv, out, qkvws);
    // 2048 waves (one per (b,h,n-tile)), 8 waves/block
    attn_wmma<<<256, 256, 0, stream>>>(qkvws, attnws);
    // 8192 threads over (b,p)
    proj_wo<<<32, 256, 0, stream>>>(attnws, Wo, bo, out);
}